// DINOLSSFPN_61435212202116
// MI455X (gfx1250) — compile-verified
//
#include <hip/hip_runtime.h>
#include <hip/hip_bf16.h>

typedef __attribute__((ext_vector_type(2))) float v2f;
typedef __attribute__((ext_vector_type(4))) float f4;
typedef __attribute__((ext_vector_type(8))) float v8f;

#define NVIEW 6
#define FHD   16      // downsampled height
#define FWD   44      // downsampled width
#define IMG_W 704
#define CCH   80
#define DBINS 112
#define GNX   128
#define GNY   128
#define NPIX  (FHD*FWD)          // 704 pixels per view (coincidentally == IMG_W)
#define WS_MAT_FLOATS (NVIEW*32) // per view: R[16] + invIda[16]

// ---------------- 4x4 helpers (tiny, runs on 6 threads once) ----------------
__device__ void inv4x4(const float* m, float* out) {
  float a[4][8];
  for (int r = 0; r < 4; ++r)
    for (int c = 0; c < 4; ++c) { a[r][c] = m[r*4+c]; a[r][c+4] = (r == c) ? 1.f : 0.f; }
  for (int col = 0; col < 4; ++col) {
    int piv = col; float best = fabsf(a[col][col]);
    for (int r = col+1; r < 4; ++r) { float v = fabsf(a[r][col]); if (v > best) { best = v; piv = r; } }
    if (piv != col)
      for (int c = 0; c < 8; ++c) { float t = a[col][c]; a[col][c] = a[piv][c]; a[piv][c] = t; }
    float rinv = 1.f / a[col][col];
    for (int c = 0; c < 8; ++c) a[col][c] *= rinv;
    for (int r = 0; r < 4; ++r) {
      if (r == col) continue;
      float f = a[r][col];
      for (int c = 0; c < 8; ++c) a[r][c] -= f * a[col][c];
    }
  }
  for (int r = 0; r < 4; ++r)
    for (int c = 0; c < 4; ++c) out[r*4+c] = a[r][c+4];
}

__device__ void mm4(const float* A, const float* B, float* C) {
  for (int i = 0; i < 4; ++i)
    for (int j = 0; j < 4; ++j) {
      float s = 0.f;
      for (int k = 0; k < 4; ++k) s += A[i*4+k] * B[k*4+j];
      C[i*4+j] = s;
    }
}

// ws[v*32 +  0..15] = R    = bda @ s2e @ inv(intrin)   (row major)
// ws[v*32 + 16..31] = inv(ida)                         (row major)
__global__ void lss_prep_mats(const float* __restrict__ s2e, const float* __restrict__ intrin,
                              const float* __restrict__ ida, const float* __restrict__ bda,
                              float* __restrict__ wsf) {
  int v = threadIdx.x;
  if (v >= NVIEW) return;
  float invI[16], comb[16], R[16], invA[16];
  inv4x4(intrin + v*16, invI);
  mm4(s2e + v*16, invI, comb);
  mm4(bda, comb, R);
  inv4x4(ida + v*16, invA);
  for (int i = 0; i < 16; ++i) { wsf[v*32 + i] = R[i]; wsf[v*32 + 16 + i] = invA[i]; }
}

// ---------------- zero the BEV accumulator (re-run every launch) -------------
__global__ void lss_zero_out(float* __restrict__ out) {
  int i = blockIdx.x * blockDim.x + threadIdx.x;   // 1280*256 == 1310720/4 exactly
  f4 z = {0.f, 0.f, 0.f, 0.f};
  reinterpret_cast<f4*>(out)[i] = z;
}

// ---------------- per-pixel 16x16 depth-min reduction ------------------------
__global__ void lss_depth_min(const float* __restrict__ lidar, float* __restrict__ dmin) {
  __shared__ float sm[256];
  int pid = blockIdx.x;                // v*NPIX + h*FWD + w
  int v  = pid / NPIX;
  int hw = pid % NPIX;
  int h = hw / FWD, w = hw % FWD;
  int t = threadIdx.x;
  int i = t >> 4, j = t & 15;
  float val = lidar[(size_t)(v*256 + h*16 + i) * IMG_W + (w*16 + j)];
  val = (val == 0.f) ? 1e5f : val;
  sm[t] = val;
  __syncthreads();
  for (int s = 128; s > 0; s >>= 1) {
    if (t < s) sm[t] = fminf(sm[t], sm[t + s]);
    __syncthreads();
  }
  if (t == 0) dmin[pid] = sm[0];
}

// ---------------- WMMA geometry + sparse splat -------------------------------
// One wave32 per tile of 16 pixels of one view. Geometry for the 16 pixels of
// each depth bin is done with a single V_WMMA_F32_16X16X4_F32:
//   D[16x16] = A[16x4] * B[4x16],  A rows 0..3 = R,  B cols = homogeneous pts.
// D layout: lane n (0..15), VGPR m -> component m of point n.
__global__ void __launch_bounds__(32)
lss_splat(const float* __restrict__ feat, const float* __restrict__ wsf,
          float* __restrict__ out) {
  const int lane = threadIdx.x;                 // 0..31
  const int v    = blockIdx.x / (NPIX / 16);    // 44 tiles per view
  const int tile = blockIdx.x % (NPIX / 16);
  const int p    = lane & 15;                   // pixel within tile (lanes l and l+16 share pixel)
  const int pix  = tile * 16 + p;
  const int h = pix / FWD, w = pix % FWD;

  const float* R    = wsf + v*32;
  const float* invA = wsf + v*32 + 16;
  const float* dbuf = wsf + WS_MAT_FLOATS;

  // soft one-hot depth weights (at most 2 non-zero bins per pixel)
  float dmin  = dbuf[v*NPIX + pix];
  float valid = (dmin < 1e5f) ? 1.f : 0.f;
  float pos = (dmin - 2.0f) * 2.0f;             // (d - d_min) / d_int
  pos = fminf(fmaxf(pos, 0.f), 111.99999f);
  int lower = (int)floorf(pos);
  lower = min(max(lower, 0), DBINS - 1);
  int upper = min(lower + 1, DBINS - 1);
  float wu = fminf(fmaxf(pos - (float)lower, 0.f), 1.f);
  float wl = (1.f - wu) * valid;
  wu *= valid;

  // frustum image-plane coordinates: linspace(0,703,44) / linspace(0,255,16)
  const float ximg = (float)w * (703.0f / 43.0f);
  const float yimg = (float)h * 17.0f;

  // A operand: lane = M row, VGPR0/1 = K{0,1} (lanes 0-15) or K{2,3} (lanes 16-31)
  const int m  = lane & 15;
  const int kb = (lane < 16) ? 0 : 2;
  v2f aop;
  aop.x = (m < 4) ? R[m*4 + kb]     : 0.f;
  aop.y = (m < 4) ? R[m*4 + kb + 1] : 0.f;

  const int   bins[2] = { lower, upper };
  const float wgts[2] = { wl, wu };

  for (int bi = 0; bi < 2; ++bi) {
    const float dv = 2.0f + 0.5f * (float)bins[bi];
    // q = inv(ida) @ [x, y, d, 1] ; then scale xy by z (the lift's nonlinearity)
    float q0 = invA[ 0]*ximg + invA[ 1]*yimg + invA[ 2]*dv + invA[ 3];
    float q1 = invA[ 4]*ximg + invA[ 5]*yimg + invA[ 6]*dv + invA[ 7];
    float q2 = invA[ 8]*ximg + invA[ 9]*yimg + invA[10]*dv + invA[11];
    float q3 = invA[12]*ximg + invA[13]*yimg + invA[14]*dv + invA[15];
    float qp0 = q0 * q2, qp1 = q1 * q2;

    // B operand: lane = N column; VGPR0/1 = K{0,1} (lanes 0-15) or K{2,3} (lanes 16-31)
    v2f bop;
    bop.x = (lane < 16) ? qp0 : q2;
    bop.y = (lane < 16) ? qp1 : q3;

    v8f acc = {0.f, 0.f, 0.f, 0.f, 0.f, 0.f, 0.f, 0.f};
    v8f dres = __builtin_amdgcn_wmma_f32_16x16x4_f32(
        false, aop, false, bop, (short)0, acc, false, false);

    // lanes 0..15: VGPR m holds component m of this lane's point
    float gx = dres[0], gy = dres[1], gz = dres[2];
    int cx = (int)floorf((gx + 51.2f) * 1.25f);   // / 0.8
    int cy = (int)floorf((gy + 51.2f) * 1.25f);
    int cz = (int)floorf((gz +  5.0f) * 0.125f);  // / 8.0
    int ok  = (cx >= 0 && cx < GNX && cy >= 0 && cy < GNY && cz == 0) ? 1 : 0;
    int idx = cy * GNX + cx;
    float wgt = wgts[bi];

    // wave-uniform broadcast per pixel; all 32 lanes split the 80 channels
    for (int pp = 0; pp < 16; ++pp) {
      int   okp  = __shfl(ok,  pp, 32);
      float wp   = __shfl(wgt, pp, 32);
      int   idxp = __shfl(idx, pp, 32);
      if (okp && wp > 0.f) {
        int pixp = tile * 16 + pp;
        int hp = pixp / FWD, wwp = pixp % FWD;
        const float* fp = feat + (size_t)v * CCH * NPIX + hp * FWD + wwp;
        float* op = out + idxp;
        for (int c = lane; c < CCH; c += 32)
          atomicAdd(op + (size_t)c * (GNX * GNY), wp * fp[(size_t)c * NPIX]);
      }
    }
  }
}

extern "C" void kernel_launch(void* const* d_in, const int* in_sizes, int n_in,
                              void* d_out, int out_size, void* d_ws, size_t ws_size,
                              hipStream_t stream) {
  const float* lidar  = (const float*)d_in[0];
  const float* feat   = (const float*)d_in[1];
  const float* s2e    = (const float*)d_in[2];
  const float* intrin = (const float*)d_in[3];
  const float* ida    = (const float*)d_in[4];
  const float* bda    = (const float*)d_in[5];
  float* out = (float*)d_out;
  float* wsf = (float*)d_ws;

  lss_prep_mats<<<1, 32, 0, stream>>>(s2e, intrin, ida, bda, wsf);
  lss_zero_out<<<1280, 256, 0, stream>>>(out);                         // 1,310,720 floats
  lss_depth_min<<<NVIEW * NPIX, 256, 0, stream>>>(lidar, wsf + WS_MAT_FLOATS);
  lss_splat<<<NVIEW * (NPIX / 16), 32, 0, stream>>>(feat, wsf, out);
}